// DTWLoss_64166811402657
// MI455X (gfx1250) — compile-verified
//
#include <hip/hip_runtime.h>
#include <math.h>

typedef float v2f __attribute__((ext_vector_type(2)));
typedef float v8f __attribute__((ext_vector_type(8)));

#define NBATCH 8
#define SEQLEN 2048
#define NFEAT  8
#define NTILES (SEQLEN / 16)   // 128

#define LOG2E_F    1.4426950408889634f
#define TWOLOG2E_F 2.8853900817779268f

// One wave32 per (batch, 16-row i-tile).
// pt tile via two chained V_WMMA_F32_16X16X4_F32 (K = 8 = 2*4).
// A is pre-scaled by 2*log2e and t2 by log2e, so the per-element exp is a
// bare v_exp_f32 (native exp2) with argument log2e*(2pt - t2_j) <= log2e*p2_i.
// Accumulate S'_r = sum_j exp(2pt - t2_j); since d_ij >= 0 the sum is bounded
// by 2048*e^{p2_r} (p2 ~ chi2(8): overflow needs p2 > ~81, P ~ 1e-13 -> safe).
// Row result: min_dist_r = p2_r - ln(S'_r); answer = mean over all rows.
__global__ __launch_bounds__(256) void dtw_softmin_wmma_kernel(
    const float* __restrict__ pred,
    const float* __restrict__ target,
    float* __restrict__ out)
{
    const int lane = threadIdx.x & 31;
    const int wave = blockIdx.x * 8 + (threadIdx.x >> 5);  // 0..1023
    const int b    = wave >> 7;                            // batch 0..7
    const int i0   = (wave & (NTILES - 1)) * 16;           // i-tile base row

    const int hv = lane >> 4;   // 0: lanes 0-15, 1: lanes 16-31
    const int n  = lane & 15;   // row-in-tile (A) / col-in-tile (B,C,D)

    // ---- A fragment: pred rows i0..i0+15, stationary across j loop ----
    // ISA 32-bit A 16x4 layout: lanes 0-15 -> {K0,K1}, lanes 16-31 -> {K2,K3}.
    const float4* pr = reinterpret_cast<const float4*>(
        pred + ((size_t)(b * SEQLEN + i0 + n)) * NFEAT);
    float4 p0 = pr[0];
    float4 p1 = pr[1];
    float p2lane = p0.x*p0.x + p0.y*p0.y + p0.z*p0.z + p0.w*p0.w
                 + p1.x*p1.x + p1.y*p1.y + p1.z*p1.z + p1.w*p1.w;
    v2f a1, a2;                              // scaled: accumulator = log2e*2pt
    a1.x = TWOLOG2E_F * (hv ? p0.z : p0.x);  // K = 2*hv
    a1.y = TWOLOG2E_F * (hv ? p0.w : p0.y);  // K = 2*hv + 1
    a2.x = TWOLOG2E_F * (hv ? p1.z : p1.x);  // K = 4 + 2*hv
    a2.y = TWOLOG2E_F * (hv ? p1.w : p1.y);  // K = 5 + 2*hv

    // p2 (natural units) of the 8 rows this lane's C VGPRs belong to
    // (row = v + 8*hv); p2 of row r lives in lanes r and r+16.
    float p2row[8];
    #pragma unroll
    for (int v = 0; v < 8; ++v)
        p2row[v] = __shfl(p2lane, v + 8 * hv, 32);

    float S[8];
    #pragma unroll
    for (int v = 0; v < 8; ++v) S[v] = 0.0f;

    const float4* tbase = reinterpret_cast<const float4*>(
        target + ((size_t)b * SEQLEN) * NFEAT);

    #pragma unroll 2
    for (int jt = 0; jt < NTILES; ++jt) {
        const int j0 = jt * 16;
        // B fragment: target row (j0+n) supplies column n; lanes 0-15 hold
        // K rows {0,1}, lanes 16-31 hold {2,3} (mirrors A layout).
        float4 t0 = tbase[(size_t)(j0 + n) * 2 + 0];
        float4 t1 = tbase[(size_t)(j0 + n) * 2 + 1];
        float t2l2 = LOG2E_F *
                   ( t0.x*t0.x + t0.y*t0.y + t0.z*t0.z + t0.w*t0.w
                   + t1.x*t1.x + t1.y*t1.y + t1.z*t1.z + t1.w*t1.w );
        v2f b1, b2;
        b1.x = hv ? t0.z : t0.x;
        b1.y = hv ? t0.w : t0.y;
        b2.x = hv ? t1.z : t1.x;
        b2.y = hv ? t1.w : t1.y;

        v8f c = {};
        c = __builtin_amdgcn_wmma_f32_16x16x4_f32(
                false, a1, false, b1, (short)0, c, false, false);
        c = __builtin_amdgcn_wmma_f32_16x16x4_f32(
                false, a2, false, b2, (short)0, c, false, false);

        // exp2(log2e*(2pt - t2)) = exp(2pt - t2): sub + v_exp + add per elem.
        #pragma unroll
        for (int v = 0; v < 8; ++v) {
            S[v] += __builtin_amdgcn_exp2f(c[v] - t2l2);
        }
    }

    // Sum S across the 16 lanes of each half-wave (columns of the row).
    #pragma unroll
    for (int off = 1; off < 16; off <<= 1) {
        #pragma unroll
        for (int v = 0; v < 8; ++v)
            S[v] += __shfl_xor(S[v], off, 32);
    }

    // min_dist_r = p2_r - ln(S'_r); lane 0 holds rows 0-7, lane 16 rows 8-15.
    float rowsum = 0.0f;
    #pragma unroll
    for (int v = 0; v < 8; ++v) rowsum += p2row[v] - __logf(S[v]);

    float val = (n == 0 ? rowsum : 0.0f);
    #pragma unroll
    for (int off = 1; off < 32; off <<= 1) val += __shfl_xor(val, off, 32);

    if (lane == 0) {
        atomicAdd(out, val * (1.0f / (float)(NBATCH * SEQLEN)));
    }
}

extern "C" void kernel_launch(void* const* d_in, const int* in_sizes, int n_in,
                              void* d_out, int out_size, void* d_ws, size_t ws_size,
                              hipStream_t stream) {
    (void)in_sizes; (void)n_in; (void)out_size; (void)d_ws; (void)ws_size;
    const float* pred   = (const float*)d_in[0];
    const float* target = (const float*)d_in[1];
    float* out = (float*)d_out;

    (void)hipMemsetAsync(out, 0, sizeof(float), stream);
    dim3 grid(128);    // 128 blocks * 8 waves = 1024 waves = 8 batches * 128 i-tiles
    dim3 block(256);
    dtw_softmin_wmma_kernel<<<grid, block, 0, stream>>>(pred, target, out);
}